// HausdorffLoss_4183298147149
// MI455X (gfx1250) — compile-verified
//
#include <hip/hip_runtime.h>
#include <hip/hip_bf16.h>
#include <math.h>

// ---------------------------------------------------------------------------
// Problem constants (from the reference)
// ---------------------------------------------------------------------------
#define B_     16
#define H_     512
#define W_     512
#define N_     (B_ * H_ * W_)        // 4,194,304 pixels
#define HW_    (H_ * W_)             // 262,144
#define ITERS_ 128                   // max(10, min(512*0.25, 200))

// Padded distance maps: 1-pixel "+inf" ring per image -> SAME padding for min-pool
#define PW_    (W_ + 2)              // 514
#define PP_    (PW_ * PW_)           // 264,196 per image
#define PTOT_  (B_ * PP_)            // 4,227,136 u32 elements

// DT tile
#define TX_    64
#define TY_    16
#define TLW_   (TX_ + 2)             // 66
#define TLH_   (TY_ + 2)             // 18

#define NBINS_ (130 * 130)           // (a in {0..128,1024->129}) x (b likewise)

typedef unsigned int  u32;
typedef unsigned short us2 __attribute__((ext_vector_type(2)));
typedef unsigned int  v4u __attribute__((ext_vector_type(4)));
typedef int           v4i __attribute__((ext_vector_type(4)));
typedef int           v8i __attribute__((ext_vector_type(8)));
typedef float         v2f __attribute__((ext_vector_type(2)));
typedef float         v8f __attribute__((ext_vector_type(8)));

static __device__ __forceinline__ us2 as_us2(u32 v) {
    union { u32 u; us2 s; } x; x.u = v; return x.s;
}
static __device__ __forceinline__ u32 as_u32(us2 v) {
    union { u32 u; us2 s; } x; x.s = v; return x.u;
}
static __device__ __forceinline__ us2 min2(us2 a, us2 b) {
    return __builtin_elementwise_min(a, b);
}

// ---------------------------------------------------------------------------
// Kernel 1: init packed distance maps (both ping-pong buffers, incl. pad ring)
//   low u16  = gt distance  (0 where y_true>0, else 1024)
//   high u16 = bg distance  (0 where y_true==0, else 1024)
//   pad ring = 0x7FFF per half (acts as +inf; never overflows on +1)
// ---------------------------------------------------------------------------
__global__ __launch_bounds__(256) void k_init(const int* __restrict__ yt,
                                              u32* __restrict__ A,
                                              u32* __restrict__ Bb) {
    int i = blockIdx.x * 256 + threadIdx.x;
    if (i >= PTOT_) return;
    int b  = i / PP_;
    int r  = i - b * PP_;
    int yy = r / PW_;
    int xx = r - yy * PW_;
    u32 v;
    if (yy == 0 || yy == PW_ - 1 || xx == 0 || xx == PW_ - 1) {
        v = 0x7FFF7FFFu;
    } else {
        int g = yt[b * HW_ + (yy - 1) * W_ + (xx - 1)];
        v = (g > 0) ? 0x04000000u   // gt=0, bg=1024
                    : 0x00000400u;  // gt=1024, bg=0
    }
    A[i] = v;
    Bb[i] = v;
}

// ---------------------------------------------------------------------------
// Kernel 2: one chamfer DT step (both DTs fused, packed u16x2)
//   d_new = min(d, minpool3x3(d) + 1)    (fg pixels keep 0 automatically)
// Tile (66x18 u32) is fetched into LDS by the Tensor Data Mover; each thread
// produces 4 consecutive rows via a column-min sliding window (18 LDS reads
// instead of 36 per thread).
// ---------------------------------------------------------------------------
__global__ __launch_bounds__(256) void k_dt(const u32* __restrict__ src,
                                            u32* __restrict__ dst) {
    __shared__ u32 tile[TLH_ * TLW_];

    const int b  = blockIdx.z;
    const int x0 = blockIdx.x * TX_;   // padded coords of tile origin
    const int y0 = blockIdx.y * TY_;

    if (threadIdx.x < 32) {
        // ---- Build Tensor DMA Descriptor (D#), see CDNA5 ISA ----
        unsigned long long gaddr =
            (unsigned long long)(uintptr_t)src +
            ((unsigned long long)(b * PP_ + y0 * PW_ + x0)) * 4ull;
        u32 ldsoff = (u32)__builtin_amdgcn_readfirstlane((int)(u32)(uintptr_t)tile);
        u32 ga_lo  = (u32)__builtin_amdgcn_readfirstlane((int)(u32)(gaddr & 0xFFFFFFFFull));
        u32 ga_hi  = (u32)__builtin_amdgcn_readfirstlane((int)(u32)(gaddr >> 32));

        // Group0: count=1 | lds_addr | global_addr[56:0] | type=2
        v4u g0 = { 1u,
                   ldsoff,
                   ga_lo,
                   ((ga_hi & 0x01FFFFFFu) | 0x80000000u) };

        // Group1: data_size=4B (code 2), tile 66x18, tensor dims big enough
        // to never clip (tiles are always in-bounds of the padded buffer),
        // row stride 514 elements.
        v8i g1 = { (int)0x00020000u,            // data_size=2 (4B), mask=0
                   (int)(514u << 16),           // tensor_dim0[15:0]=514
                   (int)(8224u << 16),          // tensor_dim1[15:0]=8224 (514*16)
                   (int)(66u  << 16),           // tile_dim0=66
                   (int)18,                     // tile_dim1=18
                   (int)514,                    // tensor_dim0_stride=514
                   0, 0 };
        v4i gz4 = { 0, 0, 0, 0 };               // 2D tensor: groups 2/3 unused
        v8i gz8 = { 0, 0, 0, 0, 0, 0, 0, 0 };

        __builtin_amdgcn_tensor_load_to_lds(g0, g1, gz4, gz4, gz8, 0);
        __builtin_amdgcn_s_wait_tensorcnt(0);
    }
    __syncthreads();

    const int lx  = threadIdx.x & (TX_ - 1);
    const int lyb = (threadIdx.x >> 6) << 2;   // 0,4,8,12 (4 consecutive rows)

    // Column mins over 3 columns for the 6 input rows feeding 4 outputs.
    us2 hmin[6], mid[6];
    #pragma unroll
    for (int i = 0; i < 6; ++i) {
        const u32* row = &tile[(lyb + i) * TLW_ + lx];
        us2 a  = as_us2(row[0]);
        us2 bq = as_us2(row[1]);
        us2 cq = as_us2(row[2]);
        mid[i]  = bq;
        hmin[i] = min2(min2(a, bq), cq);
    }

    const us2 one = { 1, 1 };
    #pragma unroll
    for (int rr = 0; rr < 4; ++rr) {
        us2 m   = min2(min2(hmin[rr], hmin[rr + 1]), hmin[rr + 2]);
        us2 res = min2(mid[rr + 1], m + one);   // center included in min-pool
        dst[b * PP_ + (y0 + 1 + lyb + rr) * PW_ + (x0 + 1 + lx)] = as_u32(res);
    }
}

// ---------------------------------------------------------------------------
// Kernel 3: zero histogram
// ---------------------------------------------------------------------------
__global__ __launch_bounds__(256) void k_zero(u32* __restrict__ hist) {
    int i = blockIdx.x * 256 + threadIdx.x;
    if (i < NBINS_) hist[i] = 0u;
}

// ---------------------------------------------------------------------------
// Kernel 4: histogram of (gt_dist, bg_dist) pairs (130x130 bins)
// ---------------------------------------------------------------------------
__global__ __launch_bounds__(256) void k_hist(const u32* __restrict__ fin,
                                              u32* __restrict__ hist) {
    int i = blockIdx.x * 256 + threadIdx.x;   // 0..N_-1
    int b = i >> 18;
    int r = i & 0x3FFFF;
    int y = r >> 9;
    int x = r & 511;
    u32 v  = fin[b * PP_ + (y + 1) * PW_ + (x + 1)];
    u32 a  = v & 0xFFFFu;
    u32 bb = v >> 16;
    u32 ia = (a  > 128u) ? 129u : a;
    u32 ib = (bb > 128u) ? 129u : bb;
    atomicAdd(&hist[ia * 130u + ib], 1u);
}

// ---------------------------------------------------------------------------
// Kernel 5: exact 99th-percentile via binary search over weight values
// (single block). weight(bin i,j) = v(i)^2 + v(j)^2, v(k)=k<=128?k:1024.
// ---------------------------------------------------------------------------
__device__ __forceinline__ int bin_weight(int idx) {
    int i = idx / 130, j = idx - i * 130;
    int vi = (i > 128) ? 1024 : i;
    int vj = (j > 128) ? 1024 : j;
    return vi * vi + vj * vj;
}

__global__ __launch_bounds__(256) void k_quant(const u32* __restrict__ hist,
                                               float* __restrict__ normp) {
    __shared__ u32 red[256];
    const double h  = 0.99 * (double)(N_ - 1);
    const long long i0 = (long long)h;
    const double frac  = h - (double)i0;

    int sel[2];
    #pragma unroll
    for (int s = 0; s < 2; ++s) {
        const u32 kp1 = (u32)(i0 + s) + 1u;   // need count(<=w) >= rank+1
        int lo = 0, hi = 2 * 1024 * 1024;
        while (lo < hi) {
            int mid = (lo + hi) >> 1;
            u32 c = 0;
            for (int bidx = threadIdx.x; bidx < NBINS_; bidx += 256)
                if (bin_weight(bidx) <= mid) c += hist[bidx];
            red[threadIdx.x] = c;
            __syncthreads();
            for (int o = 128; o > 0; o >>= 1) {
                if (threadIdx.x < o) red[threadIdx.x] += red[threadIdx.x + o];
                __syncthreads();
            }
            u32 tot = red[0];
            __syncthreads();
            if (tot >= kp1) hi = mid; else lo = mid + 1;
        }
        sel[s] = lo;
    }
    if (threadIdx.x == 0) {
        double q = (double)sel[0] + frac * (double)(sel[1] - sel[0]);
        double nrm = q > 1.0 ? q : 1.0;
        normp[0] = (float)nrm;
    }
}

// ---------------------------------------------------------------------------
// Kernel 6: weighted-MSE partial sums, reduced through V_WMMA_F32_16X16X4_F32
// (A = 64 loss terms per wave, B = ones -> tensor-core summation).
// 1024 blocks x 256 threads x 2 elems x 8 trips = N_ exactly; EXEC stays all-1.
// ---------------------------------------------------------------------------
__device__ __forceinline__ float loss_term(int i,
                                           const float* __restrict__ logits,
                                           const int* __restrict__ yt,
                                           const u32* __restrict__ fin) {
    int b = i >> 18;
    int r = i & 0x3FFFF;
    int y = r >> 9;
    int x = r & 511;
    float l0 = logits[b * 2 * HW_ + r];
    float l1 = logits[b * 2 * HW_ + HW_ + r];
    // softmax channel 1 = 1/(1+exp(l0-l1)); single v_rcp_f32, not IEEE divide
    float p  = __builtin_amdgcn_rcpf(1.0f + __expf(l0 - l1));
    float g  = (yt[i] > 0) ? 1.0f : 0.0f;
    u32 v  = fin[b * PP_ + (y + 1) * PW_ + (x + 1)];
    u32 a  = v & 0xFFFFu;
    u32 bb = v >> 16;
    float w  = (float)(a * a + bb * bb);
    float d  = p - g;
    return d * d * w;
}

__global__ __launch_bounds__(256) void k_loss(const float* __restrict__ logits,
                                              const int* __restrict__ yt,
                                              const u32* __restrict__ fin,
                                              float* __restrict__ partials) {
    v8f acc = { 0.f, 0.f, 0.f, 0.f, 0.f, 0.f, 0.f, 0.f };
    const v2f ones = { 1.0f, 1.0f };
    const int gtid = blockIdx.x * 256 + (int)threadIdx.x;

    #pragma unroll
    for (int it = 0; it < 8; ++it) {
        int i0 = it * (1024 * 256 * 2) + gtid * 2;
        v2f a;
        a.x = loss_term(i0,     logits, yt, fin);
        a.y = loss_term(i0 + 1, logits, yt, fin);
        // D = A(16x4) * ones(4x16) + C : accumulates 64 wave values into C
        acc = __builtin_amdgcn_wmma_f32_16x16x4_f32(
                  false, a, false, ones, (short)0, acc, false, false);
    }

    float s = acc[0] + acc[1] + acc[2] + acc[3] +
              acc[4] + acc[5] + acc[6] + acc[7];
    s += __shfl_xor(s, 16, 32);        // join M=0..7 / M=8..15 halves

    __shared__ float wsum[8];
    int wave = threadIdx.x >> 5;
    if ((threadIdx.x & 31) == 0) wsum[wave] = s;
    __syncthreads();
    if (threadIdx.x == 0) {
        float t = 0.f;
        #pragma unroll
        for (int w = 0; w < 8; ++w) t += wsum[w];
        partials[blockIdx.x] = t;
    }
}

// ---------------------------------------------------------------------------
// Kernel 7: final deterministic reduction + normalization
// ---------------------------------------------------------------------------
__global__ __launch_bounds__(256) void k_final(const float* __restrict__ partials,
                                               const float* __restrict__ normp,
                                               float* __restrict__ out) {
    __shared__ float red[256];
    float s = 0.f;
    for (int i = threadIdx.x; i < 1024; i += 256) s += partials[i];
    red[threadIdx.x] = s;
    __syncthreads();
    for (int o = 128; o > 0; o >>= 1) {
        if (threadIdx.x < o) red[threadIdx.x] += red[threadIdx.x + o];
        __syncthreads();
    }
    if (threadIdx.x == 0)
        out[0] = red[0] / (normp[0] * (float)N_);
}

// ---------------------------------------------------------------------------
// Host orchestration
// ---------------------------------------------------------------------------
extern "C" void kernel_launch(void* const* d_in, const int* in_sizes, int n_in,
                              void* d_out, int out_size, void* d_ws, size_t ws_size,
                              hipStream_t stream) {
    const float* logits = (const float*)d_in[0];
    const int*   y_true = (const int*)d_in[1];
    float*       out    = (float*)d_out;

    const size_t PBYTES = (size_t)PTOT_ * 4u;
    uintptr_t p = (uintptr_t)d_ws;
    auto al = [](uintptr_t v) { return (v + 255u) & ~(uintptr_t)255u; };
    u32*   bufA     = (u32*)p;   p = al(p + PBYTES);
    u32*   bufB     = (u32*)p;   p = al(p + PBYTES);
    u32*   hist     = (u32*)p;   p = al(p + (size_t)NBINS_ * 4u);
    float* normp    = (float*)p; p = al(p + 256u);
    float* partials = (float*)p;

    // 1) init packed distance maps (both buffers, including pad ring)
    k_init<<<(PTOT_ + 255) / 256, 256, 0, stream>>>(y_true, bufA, bufB);

    // 2) 128 fused chamfer-DT steps, ping-pong (L2-resident working set)
    u32* src = bufA;
    u32* dst = bufB;
    dim3 dtGrid(W_ / TX_, H_ / TY_, B_);   // 8 x 32 x 16
    for (int it = 0; it < ITERS_; ++it) {
        k_dt<<<dtGrid, 256, 0, stream>>>(src, dst);
        u32* t = src; src = dst; dst = t;
    }
    u32* fin = src;   // after an even number of swaps, result is in bufA

    // 3) histogram of (gt_dist, bg_dist) pairs
    k_zero<<<(NBINS_ + 255) / 256, 256, 0, stream>>>(hist);
    k_hist<<<N_ / 256, 256, 0, stream>>>(fin, hist);

    // 4) exact 99th percentile -> norm
    k_quant<<<1, 256, 0, stream>>>(hist, normp);

    // 5) weighted MSE partial sums (WMMA reduction), then final scalar
    k_loss<<<1024, 256, 0, stream>>>(logits, y_true, fin, partials);
    k_final<<<1, 256, 0, stream>>>(partials, normp, out);
}